// ACDC_35038343201373
// MI455X (gfx1250) — compile-verified
//
#include <hip/hip_runtime.h>
#include <hip/hip_bf16.h>
#include <stdint.h>

// Fused ACDC layer for MI455X (gfx1250, wave32):
//   out[b,c,s] = ( ifft_C( fft_C( x[b,:,s] * A ) * D )[perm[c]] + bias[perm[c]] ) / sqrt(C)
// C = 1024 = 16 * 16 * 4. Radix-16 stages = 16x16 complex DFT matmuls on
// v_wmma_f32_16x16x4_f32 (K=16 via 4 chained WMMAs); radix-4 + spectral scale
// is a fused VALU pass. Working set (1024 ch x 32 cols, stride-33 padded)
// lives in LDS. x-tile ingest uses the Tensor Data Mover: one
// tensor_load_to_lds per wave (128 rows each) with pad_interval=32dw /
// pad_amount=1dw reproducing the stride-33 LDS layout in hardware.
// Memory-bound: ~396MB traffic -> ~17us floor at 23.3 TB/s.

typedef float v2f __attribute__((ext_vector_type(2)));
typedef float v8f __attribute__((ext_vector_type(8)));
typedef unsigned int u32x4 __attribute__((ext_vector_type(4)));
typedef unsigned int u32x8 __attribute__((ext_vector_type(8)));

#define CD    1024
#define SD    4096
#define LSTR  33          // padded LDS row stride (floats)
#define PI_F  3.14159265358979323846f

union U8 { v8f v; float f[8]; };

__device__ __forceinline__ v8f wmma4(v2f a, v2f b, v8f c) {
  // D(16x16,f32) = A(16x4,f32) x B(4x16,f32) + C
  return __builtin_amdgcn_wmma_f32_16x16x4_f32(false, a, false, b, (short)0, c,
                                               false, false);
}

// One radix-16 matmul stage over the 1024x32 LDS working set.
// STAGE==1: groups g = n2 in [0,64), channel(q) = q*64 + g, twiddle tw[(m*g)&1023]
// STAGE==2: g = (k1,m2) = (g>>2, g&3), channel(q) = k1*64 + q*4 + m2,
//           twiddle tw[(m*m2*16)&1023]
// Forward: Y = W16 * X, then twiddle(out-row). Inverse: conj-twiddle(in-row),
// then Y = conj(W16) * X  (unnormalized adjoint; scales folded into D/bias).
// REAL_IN (stage-1 forward only): input is real x scaled by A on the fly;
// halves the WMMA count (Yr = Wr*X, Yi = Wi*X).
template<int STAGE, bool INV, bool REAL_IN>
__device__ __forceinline__ void dft16_stage(
    float* __restrict__ re, float* __restrict__ im,
    const float* __restrict__ twr, const float* __restrict__ twi,
    const float* __restrict__ ash,
    const v2f* Wr, const v2f* Wi, const v2f* nWi, int w, int l)
{
  const int lane16 = l >> 4;
  const int n = l & 15;
  for (int tt = 0; tt < 16; ++tt) {
    const int tile = w * 16 + tt;          // 128 tiles per stage, 16 per wave
    const int g = tile >> 1;
    const int colBase = (tile & 1) << 4;
    const int tmul = (STAGE == 1) ? g : ((g & 3) << 4);

    // ---- load B tiles (16x16), ISA 16x16x4 B layout per 4-K chunk ----
    v2f Xr[4], Xi[4];
#pragma unroll
    for (int c4 = 0; c4 < 4; ++c4) {
#pragma unroll
      for (int v = 0; v < 2; ++v) {
        const int q = (c4 << 2) + v + (lane16 << 1);   // K index 0..15
        int chan;
        if (STAGE == 1) chan = (q << 6) + g;
        else            chan = ((g >> 2) << 6) + (q << 2) + (g & 3);
        const int a = chan * LSTR + colBase + n;
        float xr = re[a];
        float xi = 0.0f;
        if (REAL_IN) {
          xr *= ash[chan];                             // fold diagonal A here
        } else {
          xi = im[a];
          if (INV) {                                   // conj twiddle on inputs
            const int idx = (q * tmul) & (CD - 1);
            const float tr = twr[idx], ti = twi[idx];
            const float nr = xr * tr + xi * ti;
            const float ni = xi * tr - xr * ti;
            xr = nr; xi = ni;
          }
        }
        if (v == 0) { Xr[c4].x = xr; Xi[c4].x = xi; }
        else        { Xr[c4].y = xr; Xi[c4].y = xi; }
      }
    }

    // ---- complex 16x16 matmul via chained v_wmma_f32_16x16x4_f32 ----
    U8 Yr = {}, Yi = {};
#pragma unroll
    for (int c4 = 0; c4 < 4; ++c4) {
      if (REAL_IN) {          // X real: 8 WMMAs total
        Yr.v = wmma4(Wr[c4],  Xr[c4], Yr.v);
        Yi.v = wmma4(Wi[c4],  Xr[c4], Yi.v);
      } else if (!INV) {      // W  = Wr + i*Wi
        Yr.v = wmma4(Wr[c4],  Xr[c4], Yr.v);
        Yr.v = wmma4(nWi[c4], Xi[c4], Yr.v);
        Yi.v = wmma4(Wi[c4],  Xr[c4], Yi.v);
        Yi.v = wmma4(Wr[c4],  Xi[c4], Yi.v);
      } else {                // W* = Wr - i*Wi
        Yr.v = wmma4(Wr[c4],  Xr[c4], Yr.v);
        Yr.v = wmma4(Wi[c4],  Xi[c4], Yr.v);
        Yi.v = wmma4(Wr[c4],  Xi[c4], Yi.v);
        Yi.v = wmma4(nWi[c4], Xr[c4], Yi.v);
      }
    }

    // ---- twiddle (forward) and store; C/D layout: M = r + 8*(lane>=16) ----
#pragma unroll
    for (int r = 0; r < 8; ++r) {
      const int m = r + (lane16 << 3);
      float yr = Yr.f[r], yi = Yi.f[r];
      if (!INV) {
        const int idx = (m * tmul) & (CD - 1);
        const float tr = twr[idx], ti = twi[idx];
        const float nr = yr * tr - yi * ti;
        const float ni = yr * ti + yi * tr;
        yr = nr; yi = ni;
      }
      int chan;
      if (STAGE == 1) chan = (m << 6) + g;
      else            chan = ((g >> 2) << 6) + (m << 2) + (g & 3);
      const int a = chan * LSTR + colBase + n;
      re[a] = yr;
      im[a] = yi;
    }
  }
}

__global__ void __launch_bounds__(256)
acdc_fused_fft_kernel(const float* __restrict__ x, const float* __restrict__ A,
                      const float* __restrict__ D, const float* __restrict__ bias,
                      const int* __restrict__ perm, float* __restrict__ out)
{
  extern __shared__ float smem[];
  float* re  = smem;                 // [1024 * 33]
  float* im  = re  + CD * LSTR;      // [1024 * 33]
  float* twr = im  + CD * LSTR;      // [1024]
  float* twi = twr + CD;             // [1024]
  float* ash = twi + CD;             // [1024] copy of A

  const int t = threadIdx.x;
  const int l = t & 31;
  const int w = t >> 5;
  const int lane16 = l >> 4;

  const int blk = blockIdx.x;        // 1024 blocks = 8 batches * 128 col-tiles
  const int b   = blk >> 7;
  const int s0  = (blk & 127) << 5;  // 32 spatial columns per workgroup

  // ---------------- TDM ingest of the x tile ----------------
  // Each wave DMAs 128 rows (channels) x 32 floats from global into the re[]
  // array, with hardware padding of 1 DWORD per 32 DWORDs -> stride 33 floats.
  {
    const int wu = __builtin_amdgcn_readfirstlane(w);   // uniform wave id
    const float* gsrc = x + ((size_t)b << 22) + ((size_t)(wu * 128) << 12) + s0;
    const uint64_t ga = (uint64_t)(uintptr_t)gsrc;
    const uint32_t lds_addr =
        (uint32_t)(uintptr_t)re + (uint32_t)(wu * 128 * LSTR * 4);

    u32x4 g0;
    g0[0] = 1u;                                  // count=1 (valid), user mode
    g0[1] = lds_addr;                            // LDS byte address
    g0[2] = (uint32_t)ga;                        // global_addr[31:0]
    g0[3] = ((uint32_t)(ga >> 32) & 0x01FFFFFFu) // global_addr[56:32]
            | (2u << 30);                        // type=2 ("image")

    u32x8 g1;
    g1[0] = (2u << 16)        // data_size = 4 bytes
          | (1u << 20)        // pad_enable
          | (4u << 22)        // pad_interval: 32 DWORDs
          | (0u << 25);       // pad_amount: 1 DWORD  -> row stride 33 floats
    g1[1] = (4096u & 0xFFFFu) << 16;                       // tensor_dim0 lo16
    g1[2] = ((4096u >> 16) & 0xFFFFu) | ((1024u & 0xFFFFu) << 16); // td0 hi|td1 lo
    g1[3] = ((1024u >> 16) & 0xFFFFu) | (32u << 16);       // td1 hi | tile_dim0=32
    g1[4] = 128u;                                          // tile_dim1=128, tile_dim2=0
    g1[5] = 4096u;                                         // tensor_dim0_stride lo32
    g1[6] = 0u;                                            // stride0 hi | stride1 lo
    g1[7] = 0u;                                            // stride1 hi
    u32x4 gz = {0u, 0u, 0u, 0u};                           // groups 2/3 unused (2D)

    asm volatile("tensor_load_to_lds %0, %1, %2, %3"
                 :
                 : "s"(g0), "s"(g1), "s"(gz), "s"(gz)
                 : "memory");
  }

  // Overlap with the DMA: twiddle table tw[n] = exp(-2*pi*i*n/1024) and A copy.
  for (int i = t; i < CD; i += 256) {
    const float ang = -(2.0f * PI_F / 1024.0f) * (float)i;
    twr[i] = __cosf(ang);
    twi[i] = __sinf(ang);
    ash[i] = A[i];
  }

  __builtin_amdgcn_s_wait_tensorcnt(0);   // my wave's DMA slice is in LDS
  __syncthreads();                        // everyone's slice is in LDS

  // DFT16 matrices as resident WMMA A-operands (16x4 layout: v0 holds K0|K2,
  // v1 holds K1|K3 across half-waves). f32 WMMA has no A-negate, keep -Wi too.
  v2f Wr[4], Wi[4], nWi[4];
  {
    const int M = l & 15;
#pragma unroll
    for (int c4 = 0; c4 < 4; ++c4) {
#pragma unroll
      for (int v = 0; v < 2; ++v) {
        const int K = (c4 << 2) + v + (lane16 << 1);
        const int idx = (M * K * 64) & (CD - 1);   // exp(-2*pi*i*M*K/16)
        const float cr = twr[idx];
        const float ci = twi[idx];
        if (v == 0) { Wr[c4].x = cr; Wi[c4].x = ci; nWi[c4].x = -ci; }
        else        { Wr[c4].y = cr; Wi[c4].y = ci; nWi[c4].y = -ci; }
      }
    }
  }

  dft16_stage<1, false, true >(re, im, twr, twi, ash, Wr, Wi, nWi, w, l);
  __syncthreads();
  dft16_stage<2, false, false>(re, im, twr, twi, ash, Wr, Wi, nWi, w, l);
  __syncthreads();

  // Fused: forward radix-4 -> spectral scale by D/(C*sqrt(C)) -> inverse radix-4.
  // addr = (k1*64 + j1*4 + m2)*LSTR + col ; frequency k = k1 + 16*j1 + 256*j2.
  for (int i = 0; i < 32; ++i) {
    const int p  = w + (i << 3);              // (k1,j1) pair 0..255, wave-uniform
    const int kb = (p >> 4) + ((p & 15) << 4);
    const int base = (p << 2) * LSTR + l;
    float ar = re[base],            ai = im[base];
    float br = re[base + LSTR],     bi = im[base + LSTR];
    float cr = re[base + 2 * LSTR], ci = im[base + 2 * LSTR];
    float dr = re[base + 3 * LSTR], di = im[base + 3 * LSTR];
    // forward: X[j2] = sum_m2 v[m2] * (-i)^(m2*j2)
    const float e0r = ar + cr, e0i = ai + ci, e1r = ar - cr, e1i = ai - ci;
    const float o0r = br + dr, o0i = bi + di, o1r = br - dr, o1i = bi - di;
    float X0r = e0r + o0r, X0i = e0i + o0i;
    float X2r = e0r - o0r, X2i = e0i - o0i;
    float X1r = e1r + o1i, X1i = e1i - o1r;
    float X3r = e1r - o1i, X3i = e1i + o1r;
    const float sc = 1.0f / 32768.0f;         // 1/(1024*32): ifft norm * 1/sqrt(C)
    const float d0 = D[kb]       * sc;
    const float d1 = D[kb + 256] * sc;
    const float d2 = D[kb + 512] * sc;
    const float d3 = D[kb + 768] * sc;
    X0r *= d0; X0i *= d0;  X1r *= d1; X1i *= d1;
    X2r *= d2; X2i *= d2;  X3r *= d3; X3i *= d3;
    // inverse: v[m2] = sum_j2 X[j2] * (+i)^(m2*j2)
    const float f0r = X0r + X2r, f0i = X0i + X2i, f1r = X0r - X2r, f1i = X0i - X2i;
    const float g0r = X1r + X3r, g0i = X1i + X3i, g1r = X1r - X3r, g1i = X1i - X3i;
    re[base]            = f0r + g0r; im[base]            = f0i + g0i;
    re[base + LSTR]     = f1r - g1i; im[base + LSTR]     = f1i + g1r;
    re[base + 2 * LSTR] = f0r - g0r; im[base + 2 * LSTR] = f0i - g0i;
    re[base + 3 * LSTR] = f1r + g1i; im[base + 3 * LSTR] = f1i - g1r;
  }
  __syncthreads();

  dft16_stage<2, true, false>(re, im, twr, twi, ash, Wr, Wi, nWi, w, l);
  __syncthreads();
  dft16_stage<1, true, false>(re, im, twr, twi, ash, Wr, Wi, nWi, w, l);
  __syncthreads();

  // Gather by perm, add bias/sqrt(C), write interleaved complex64 (coalesced).
  float2* out2 = (float2*)out;
  for (int c = w; c < CD; c += 8) {
    const int pc = perm[c];
    const float bsc = bias[pc] * 0.03125f;    // 1/sqrt(1024)
    const int a = pc * LSTR + l;
    float2 v2;
    v2.x = re[a] + bsc;
    v2.y = im[a];
    out2[((size_t)b << 22) + ((size_t)c << 12) + (size_t)(s0 + l)] = v2;
  }
}

extern "C" void kernel_launch(void* const* d_in, const int* in_sizes, int n_in,
                              void* d_out, int out_size, void* d_ws, size_t ws_size,
                              hipStream_t stream) {
  (void)in_sizes; (void)n_in; (void)out_size; (void)d_ws; (void)ws_size;
  const float* x    = (const float*)d_in[0];
  const float* A    = (const float*)d_in[1];
  const float* D    = (const float*)d_in[2];
  const float* bias = (const float*)d_in[3];
  const int*   perm = (const int*)d_in[4];
  float* out = (float*)d_out;

  const size_t shmem = (size_t)(2 * CD * LSTR + 3 * CD) * sizeof(float); // ~282.6KB
  dim3 grid(1024);   // (B * S / 32) = 8 * 4096 / 32
  dim3 block(256);   // 8 wave32 waves
  hipLaunchKernelGGL(acdc_fused_fft_kernel, grid, block, shmem, stream,
                     x, A, D, bias, perm, out);
}